// GRUCell_61323543052429
// MI455X (gfx1250) — compile-verified
//
#include <hip/hip_runtime.h>
#include <stdint.h>

typedef float v2f __attribute__((ext_vector_type(2)));
typedef float v8f __attribute__((ext_vector_type(8)));
typedef unsigned int v4u __attribute__((ext_vector_type(4)));
typedef int v8i __attribute__((ext_vector_type(8)));
typedef int v4i __attribute__((ext_vector_type(4)));

#define NVAR 64
#define NU   1024
#define ROWS 16
#define H_STRIDE 1028   // 1024 + 1 dword pad per 256 dwords (matches TDM pad_interval=7, pad_amount=1dw)
#define G_STRIDE 2052   // 2048 + 4 -> conflict-free sigmoid stores, bank spread on reads

#define OFF_H 0
#define OFF_G (ROWS * H_STRIDE)                 // 16448 floats
#define OFF_X (OFF_G + ROWS * G_STRIDE)         // 49280 floats
#define SMEM_FLOATS (OFF_X + ROWS * NVAR)       // 50304 floats = 201216 bytes (< 320 KB/WGP)

__device__ __forceinline__ float fast_sigmoid(float x) {
    // 1/(1+e^-x) with HW v_rcp_f32 (monotone, handles inf/0 correctly)
    return __builtin_amdgcn_rcpf(1.f + __expf(-x));
}

__device__ __forceinline__ float fast_tanh(float x) {
#if __has_builtin(__builtin_amdgcn_tanhf)
    return __builtin_amdgcn_tanhf(x);          // gfx1250 V_TANH_F32
#else
    // tanh(x) = 1 - 2/(e^{2x}+1); exp->inf/0 gives correct +/-1 limits via rcp
    return 1.f - 2.f * __builtin_amdgcn_rcpf(__expf(2.f * x) + 1.f);
#endif
}

__global__ __launch_bounds__(256)
void gru_fused_wmma(const float* __restrict__ x, const float* __restrict__ hidden,
                    const float* __restrict__ gwI, const float* __restrict__ gwH,
                    const float* __restrict__ gbias, const float* __restrict__ uwI,
                    const float* __restrict__ uwH, const float* __restrict__ ubias,
                    float* __restrict__ out)
{
    extern __shared__ float smem[];
    float* sh = smem + OFF_H;   // hidden tile [16][1028] (padded by TDM)
    float* sg = smem + OFF_G;   // sigmoid(gate) flat cols [16][2052]
    float* sx = smem + OFF_X;   // x tile [16][64]

    const int tid  = threadIdx.x;
    const int wave = tid >> 5;
    const int lane = tid & 31;
    const int m16  = lane & 15;
    const bool lo  = lane < 16;
    const int hi2  = (lane >> 4) << 1;          // K sub-offset for hi half-wave (f32 A/B frag layout)
    const int b0   = blockIdx.x * ROWS;

    // ---- stage x tile: 16 rows x 64 f32 = 1024 floats, one float4 per thread (coalesced) ----
    ((float4*)sx)[tid] = ((const float4*)(x + (size_t)b0 * NVAR))[tid];

    // ---- TDM: DMA hidden[b0:b0+16, 0:1024] (f32) -> LDS with padding (wave 0 issues) ----
    if (tid < 32) {
        uint64_t ga = (uint64_t)(uintptr_t)(hidden + (size_t)b0 * NU);
        uint32_t lds_off = (uint32_t)(uintptr_t)sh;
        v4u g0;
        g0.x = 1u;                                            // count=1, user descriptor
        g0.y = lds_off;                                       // lds_addr
        g0.z = (uint32_t)(ga & 0xFFFFFFFFu);                  // global_addr[31:0]
        g0.w = (uint32_t)((ga >> 32) & 0x01FFFFFFu) | (2u << 30); // global_addr[56:32] | type=2
        v8i g1;
        g1[0] = (int)((2u << 16) | (1u << 20) | (7u << 22));  // data_size=4B, pad_enable, interval=256dw, amount=1dw
        g1[1] = (int)((1024u & 0xFFFFu) << 16);               // tensor_dim0 lo16
        g1[2] = (int)((1024u >> 16) | (16384u << 16));        // tensor_dim0 hi16 | tensor_dim1 lo16
        g1[3] = (int)((16384u >> 16) | (1024u << 16));        // tensor_dim1 hi16 | tile_dim0=1024
        g1[4] = ROWS;                                         // tile_dim1=16, tile_dim2=0
        g1[5] = 1024;                                         // tensor_dim0_stride lo32
        g1[6] = 0;
        g1[7] = 0;
        v4i zz4 = {};
        v8i zz8 = {};
        __builtin_amdgcn_tensor_load_to_lds(g0, g1, zz4, zz4, zz8, 0);
        __builtin_amdgcn_s_wait_tensorcnt(0);
    }
    __syncthreads();

    const int vbase = wave * 8;
    const float loMask = lo ? 1.f : 0.f;        // zero A-side of augmented K rows on hi half-wave

    // ================= Pass 1: gate GEMMs, store sigmoid(G) at flat cols v*32+g =================
    #pragma unroll
    for (int vi = 0; vi < 8; ++vi) {
        const int v = vbase + vi;
        // A fragment: A[m,k] = hidden[b0+m, k*64+v]; f32 16x4 layout, chained over 4 k-groups
        v2f A[5];
        #pragma unroll
        for (int kg = 0; kg < 4; ++kg) {
            int u   = (kg * 4 + hi2) * 64 + v;            // u>>8 == kg for this range
            int idx = m16 * H_STRIDE + u + kg;            // padded LDS offset
            A[kg].x = sh[idx];
            A[kg].y = sh[idx + 64];
        }
        // Folded x*wI + bias as a 5th k=4 WMMA step: A aug = {x,1,0,0} (hi lanes zeroed),
        // B aug loaded unconditionally (A-side zeros kill the K=18/19 products).
        A[4].x = sx[m16 * NVAR + v] * loMask;
        A[4].y = loMask;

        #pragma unroll
        for (int half = 0; half < 2; ++half) {
            const int gb = half * 16;
            const float* wb = gwH + v * 512 + (gb + m16) * 16;  // W[v, g, k] row-major in k
            v2f Bf[5];
            #pragma unroll
            for (int kg = 0; kg < 4; ++kg)
                Bf[kg] = *(const v2f*)(wb + kg * 4 + hi2);      // B[k,n] = W[gb+n, k]
            Bf[4].x = gwI[v * 32 + gb + m16];
            Bf[4].y = gbias[v * 32 + gb + m16];

            v8f acc = {};
            #pragma unroll
            for (int kg = 0; kg < 5; ++kg)
                acc = __builtin_amdgcn_wmma_f32_16x16x4_f32(
                          false, A[kg], false, Bf[kg], (short)0, acc, false, false);

            #pragma unroll
            for (int j = 0; j < 8; ++j) {
                int M = j + ((lane >> 4) << 3);           // C layout: M = j (+8 for hi half)
                sg[M * G_STRIDE + v * 32 + gb + m16] = fast_sigmoid(acc[j]);
            }
        }
    }
    __syncthreads();

    // ================= Pass 2: update GEMMs with A' = h .* r, then combine =================
    #pragma unroll
    for (int vi = 0; vi < 8; ++vi) {
        const int v = vbase + vi;
        v2f A[5];
        #pragma unroll
        for (int kg = 0; kg < 4; ++kg) {
            int u    = (kg * 4 + hi2) * 64 + v;
            int hidx = m16 * H_STRIDE + u + kg;
            int ridx = m16 * G_STRIDE + 1024 + u;         // r for (v,k) is flat gate col 1024+64k+v
            A[kg].x = sh[hidx]      * sg[ridx];
            A[kg].y = sh[hidx + 64] * sg[ridx + 64];
        }
        A[4].x = sx[m16 * NVAR + v] * loMask;
        A[4].y = loMask;

        const float* wb = uwH + v * 256 + m16 * 16;
        v2f Bf[5];
        #pragma unroll
        for (int kg = 0; kg < 4; ++kg)
            Bf[kg] = *(const v2f*)(wb + kg * 4 + hi2);
        Bf[4].x = uwI[v * 16 + m16];
        Bf[4].y = ubias[v * 16 + m16];

        v8f acc = {};
        #pragma unroll
        for (int kg = 0; kg < 5; ++kg)
            acc = __builtin_amdgcn_wmma_f32_16x16x4_f32(
                      false, A[kg], false, Bf[kg], (short)0, acc, false, false);

        #pragma unroll
        for (int j = 0; j < 8; ++j) {
            int M = j + ((lane >> 4) << 3);
            int u = v * 16 + m16;                          // output unit index
            float z    = sg[M * G_STRIDE + u];             // sigma(z), flat col u < 1024
            float hold = sh[M * H_STRIDE + u + (u >> 8)];  // padded hidden read
            float g    = fast_tanh(acc[j]);
            out[(size_t)(b0 + M) * NU + u] = fmaf(z, hold - g, g);  // z*h + (1-z)*g
        }
    }
}

extern "C" void kernel_launch(void* const* d_in, const int* in_sizes, int n_in,
                              void* d_out, int out_size, void* d_ws, size_t ws_size,
                              hipStream_t stream) {
    (void)n_in; (void)out_size; (void)d_ws; (void)ws_size;
    const float* x     = (const float*)d_in[0];
    const float* h     = (const float*)d_in[1];
    const float* gwI   = (const float*)d_in[2];
    const float* gwH   = (const float*)d_in[3];
    const float* gbias = (const float*)d_in[4];
    const float* uwI   = (const float*)d_in[5];
    const float* uwH   = (const float*)d_in[6];
    const float* ubias = (const float*)d_in[7];
    float* out = (float*)d_out;

    const int B = in_sizes[0] / NVAR;          // 16384
    dim3 grid(B / ROWS), block(256);
    size_t shmem = (size_t)SMEM_FLOATS * sizeof(float);
    gru_fused_wmma<<<grid, block, shmem, stream>>>(x, h, gwI, gwH, gbias, uwI, uwH, ubias, out);
}